// MaskFunctionHaar_66657892434287
// MI455X (gfx1250) — compile-verified
//
#include <hip/hip_runtime.h>

// CDNA5 / gfx1250. Haar-feature MLP, fused single kernel.
// - W1 (32KB) staged to LDS with GLOBAL_LOAD_ASYNC_TO_LDS_B128 (ASYNCcnt path)
// - Layer2 (16 samples x 8 -> 8) via two chained V_WMMA_F32_16X16X4_F32
// - Layer1 sparse Haar gathers from LDS, layer3 tiny VALU matvec

typedef __attribute__((ext_vector_type(2))) float v2f;
typedef __attribute__((ext_vector_type(8))) float v8f;

#define LEVELS 10
#define NB     1023
#define HID    8
#define FEAT   3
#define BLOCK  256
#define WPB    (BLOCK / 32)
#define GRIDB  256

__global__ __launch_bounds__(BLOCK) void haar_mlp_kernel(
    const float* __restrict__ t,
    const float* __restrict__ W1,
    const float* __restrict__ b1,
    const float* __restrict__ W2,
    const float* __restrict__ b2,
    const float* __restrict__ W3,
    const float* __restrict__ b3,
    float* __restrict__ out,
    int ngroups)
{
    __shared__ __align__(16) float sW1[NB * HID];        // 32736 B
    __shared__ float sH2[WPB * 16 * HID];                // 4 KB, per-wave tiles

    const int tid = threadIdx.x;

    // ---- Stage 0: async W1 -> LDS (global_load_async_to_lds_b128, ASYNCcnt) ----
    {
        const float4* W1v = (const float4*)W1;
        float4* sW1v = (float4*)sW1;
        for (int i = tid; i < (NB * HID) / 4; i += BLOCK) {
            unsigned int ldsOff = (unsigned int)(uintptr_t)(&sW1v[i]); // LDS byte addr
            const void* gaddr = (const void*)(&W1v[i]);
            asm volatile("global_load_async_to_lds_b128 %0, %1, off"
                         :
                         : "v"(ldsOff), "v"(gaddr)
                         : "memory");
        }
        asm volatile("s_wait_asynccnt 0" ::: "memory");
    }

    const int lane = tid & 31;
    const int wave = tid >> 5;
    const int n    = lane & 15;     // N column (and sample id within group)
    const int hi   = lane >> 4;     // which lane-half
    const int kb   = hi * 2;        // K base within a 4-chunk (A layout)

    // ---- Wave-invariant WMMA operands (overlap with async LDS fill) ----
    // B for WMMA#1 = W2 rows 0..3 ; B for WMMA#2 = rows 4..7. N>=8 padded 0.
    v2f B1, B2;
    B1.x = (n < HID) ? W2[(kb + 0) * HID + n] : 0.0f;
    B1.y = (n < HID) ? W2[(kb + 1) * HID + n] : 0.0f;
    B2.x = (n < HID) ? W2[(kb + 4) * HID + n] : 0.0f;
    B2.y = (n < HID) ? W2[(kb + 5) * HID + n] : 0.0f;

    // C = b2 broadcast along M (same value in all 8 C VGPRs per lane)
    const float cb = (n < HID) ? b2[n] : 0.0f;
    v8f Cin;
#pragma unroll
    for (int i = 0; i < 8; ++i) Cin[i] = cb;

    // b1 components this lane accumulates (its A slots): kb, kb+1, kb+4, kb+5
    const float b1c0 = b1[kb + 0];
    const float b1c1 = b1[kb + 1];
    const float b1c2 = b1[kb + 4];
    const float b1c3 = b1[kb + 5];

    // Layer-3 weights in registers (tiny)
    float w3r[HID * FEAT];
#pragma unroll
    for (int i = 0; i < HID * FEAT; ++i) w3r[i] = W3[i];
    const float bb0 = b3[0], bb1 = b3[1], bb2v = b3[2];

    __syncthreads();   // all waves' async fills (each already waited) visible

    float* sh2 = &sH2[wave * 16 * HID];

    const int gw0     = blockIdx.x * WPB + wave;
    const int gstride = gridDim.x * WPB;

    for (int g = gw0; g < ngroups; g += gstride) {
        const int base = g * 16;

        // Prefetch next group's t chunk (speculative; lowers to global_prefetch_b8)
        __builtin_prefetch(t + (size_t)(g + gstride) * 16, 0, 0);

        // ---- Layer 1: sparse Haar gather from LDS-resident W1 ----
        const float tv = t[base + n];
        float h0 = b1c0, h1 = b1c1, h2 = b1c2, h3 = b1c3;
#pragma unroll
        for (int j = 0; j < LEVELS; ++j) {
            const float sc = tv * (float)(1 << j);
            const float fl = floorf(sc);
            const float fr = sc - fl;
            const float s  = (fr < 0.5f) ? 1.0f : -1.0f;
            const int row  = (((1 << j) - 1) + (int)fl) * HID;
            h0 = fmaf(s, sW1[row + kb + 0], h0);
            h1 = fmaf(s, sW1[row + kb + 1], h1);
            h2 = fmaf(s, sW1[row + kb + 4], h2);
            h3 = fmaf(s, sW1[row + kb + 5], h3);
        }
        // ReLU after layer 1
        h0 = fmaxf(h0, 0.0f);
        h1 = fmaxf(h1, 0.0f);
        h2 = fmaxf(h2, 0.0f);
        h3 = fmaxf(h3, 0.0f);

        v2f A1; A1.x = h0; A1.y = h1;   // K = kb, kb+1     (chunk 0..3)
        v2f A2; A2.x = h2; A2.y = h3;   // K = kb+4, kb+5   (chunk 4..7)

        // ---- Layer 2: D = A1*B1 + A2*B2 + b2 (two chained f32 WMMAs) ----
        v8f D = __builtin_amdgcn_wmma_f32_16x16x4_f32(
            false, A1, false, B1, (short)0, Cin, false, false);
        D = __builtin_amdgcn_wmma_f32_16x16x4_f32(
            false, A2, false, B2, (short)0, D, false, false);

        // ---- ReLU + spill h2 tile to per-wave LDS ----
        if (n < HID) {
            const int mhi = hi * 8;
#pragma unroll
            for (int r = 0; r < 8; ++r) {
                sh2[(mhi + r) * HID + n] = fmaxf(D[r], 0.0f);
            }
        }
        asm volatile("s_wait_dscnt 0" ::: "memory");  // within-wave LDS RAW

        // ---- Layer 3: 8 -> 3 in VALU, lanes 0..15, one sample each ----
        if (lane < 16) {
            float a0 = bb0, a1 = bb1, a2 = bb2v;
#pragma unroll
            for (int k = 0; k < HID; ++k) {
                const float hv = sh2[lane * HID + k];
                a0 = fmaf(hv, w3r[k * FEAT + 0], a0);
                a1 = fmaf(hv, w3r[k * FEAT + 1], a1);
                a2 = fmaf(hv, w3r[k * FEAT + 2], a2);
            }
            float* op = out + (size_t)(base + lane) * FEAT;
            op[0] = a0;
            op[1] = a1;
            op[2] = a2;
        }
    }
}

extern "C" void kernel_launch(void* const* d_in, const int* in_sizes, int n_in,
                              void* d_out, int out_size, void* d_ws, size_t ws_size,
                              hipStream_t stream) {
    const float* t  = (const float*)d_in[0];
    const float* W1 = (const float*)d_in[1];
    const float* b1 = (const float*)d_in[2];
    const float* W2 = (const float*)d_in[3];
    const float* b2 = (const float*)d_in[4];
    const float* W3 = (const float*)d_in[5];
    const float* b3 = (const float*)d_in[6];
    float* out = (float*)d_out;

    const int nsamp   = in_sizes[0];      // B*T = 131072
    const int ngroups = nsamp / 16;       // 8192 groups of 16 samples

    haar_mlp_kernel<<<GRIDB, BLOCK, 0, stream>>>(t, W1, b1, W2, b2, W3, b3, out, ngroups);
}